// CapsuleLinear_53996328845967
// MI455X (gfx1250) — compile-verified
//
#include <hip/hip_runtime.h>

typedef __attribute__((ext_vector_type(2))) float v2f;
typedef __attribute__((ext_vector_type(8))) float v8f;

#define I_IN   1152
#define L_IN   8
#define J_OUT  64
#define K_OUT  16
#define ITILE  128
#define NCHUNK (I_IN / ITILE)   // 9
#define NITER  3
#define EPS_F  1e-9f

// One workgroup per sample n. All routing state (x[n], w, V, logits-chunk,
// t, s/out) lives in LDS; priors are never materialized. Logits are
// recomputed each iteration as x[n] . V^T via V_WMMA_F32_16X16X4_F32.
__global__ __launch_bounds__(256)
void capsule_route_kernel(const float* __restrict__ x,
                          const float* __restrict__ w,
                          float* __restrict__ out) {
    extern __shared__ float smem[];
    float* xs = smem;                        // [1152][8]   x[n]
    float* ws = xs + I_IN * L_IN;            // [64][16][8] weight
    float* Vs = ws + J_OUT * K_OUT * L_IN;   // [64][8]     accumulated v (logit generator)
    float* lc = Vs + J_OUT * L_IN;           // [64][128]   logits/probs chunk
    float* ts = lc + J_OUT * ITILE;          // [64][8]     t = probs . x
    float* sb = ts + J_OUT * L_IN;           // [64][16]    s / out
    float* sc = sb + J_OUT * K_OUT;          // [64]        squash scale

    const int tid  = threadIdx.x;
    const int n    = blockIdx.x;
    const int lane = tid & 31;
    const int wave = tid >> 5;    // 0..7 : this wave's i-tile column
    const int m16  = lane & 15;   // row (A) / col (B,D) index within fragment
    const int half = lane >> 4;

    // ---- stage x[n] and w into LDS (vectorized) ----
    {
        const float4* x4  = (const float4*)(x + (size_t)n * I_IN * L_IN);
        float4*       xs4 = (float4*)xs;
        for (int i = tid; i < I_IN * L_IN / 4; i += 256) xs4[i] = x4[i];
        const float4* w4  = (const float4*)w;
        float4*       ws4 = (float4*)ws;
        for (int i = tid; i < J_OUT * K_OUT * L_IN / 4; i += 256) ws4[i] = w4[i];
        for (int i = tid; i < J_OUT * L_IN; i += 256) Vs[i] = 0.0f;
    }
    __syncthreads();

    // each thread owns two (j,l) accumulator slots of t: tid and tid+256
    const int j0 = tid >> 3, l0 = tid & 7;
    const int p1 = tid + 256;
    const int j1 = p1 >> 3, l1 = p1 & 7;

    for (int iter = 0; iter < NITER; ++iter) {
        float t0 = 0.0f, t1 = 0.0f;

        for (int chunk = 0; chunk < NCHUNK; ++chunk) {
            const int ibase = chunk * ITILE;

            // ---- logits chunk: (64 x 128) = V(64x8) . x_chunk^T(8x128) ----
            // 8 waves each own one 16-wide i-tile (it = wave) and sweep the
            // 4 j-tiles. B fragments (x side) are invariant across j-tiles:
            // load once, reuse 4x. K=8 -> 2 k-steps of 4.
            {
                const int xrow = (ibase + wave * 16 + m16) * L_IN + 2 * half;
                v2f b0, b1;                       // B 4x16: b[v] = B[v+2*half][n16]
                b0.x = xs[xrow + 0];
                b0.y = xs[xrow + 1];
                b1.x = xs[xrow + 4];
                b1.y = xs[xrow + 5];
#pragma unroll
                for (int jt = 0; jt < 4; ++jt) {
                    const int vrow = (jt * 16 + m16) * L_IN + 2 * half;
                    v2f a0, a1;                   // A 16x4: a[v] = A[m16][v+2*half]
                    a0.x = Vs[vrow + 0];
                    a0.y = Vs[vrow + 1];
                    a1.x = Vs[vrow + 4];
                    a1.y = Vs[vrow + 5];
                    v8f c = {0.f, 0.f, 0.f, 0.f, 0.f, 0.f, 0.f, 0.f};
                    c = __builtin_amdgcn_wmma_f32_16x16x4_f32(
                            false, a0, false, b0, (short)0, c, false, false);
                    c = __builtin_amdgcn_wmma_f32_16x16x4_f32(
                            false, a1, false, b1, (short)0, c, false, false);
                    // D 16x16: c[v] -> row jt*16 + 8*half + v, col wave*16 + m16
#pragma unroll
                    for (int v = 0; v < 8; ++v)
                        lc[(jt * 16 + 8 * half + v) * ITILE + wave * 16 + m16] = c[v];
                }
            }
            __syncthreads();

            // ---- softmax over J per i-column (in place) ----
            if (tid < ITILE) {
                float mx = -3.402823466e38f;
                for (int j = 0; j < J_OUT; ++j)
                    mx = fmaxf(mx, lc[j * ITILE + tid]);
                float sum = 0.0f;
                for (int j = 0; j < J_OUT; ++j) {
                    float e = __expf(lc[j * ITILE + tid] - mx);
                    lc[j * ITILE + tid] = e;
                    sum += e;
                }
                const float inv = 1.0f / sum;
                for (int j = 0; j < J_OUT; ++j)
                    lc[j * ITILE + tid] *= inv;
            }
            __syncthreads();

            // ---- t[j][l] += sum_i probs[j][i] * x[i][l] ----
#pragma unroll 4
            for (int ic = 0; ic < ITILE; ++ic) {
                t0 = fmaf(lc[j0 * ITILE + ic], xs[(ibase + ic) * L_IN + l0], t0);
                t1 = fmaf(lc[j1 * ITILE + ic], xs[(ibase + ic) * L_IN + l1], t1);
            }
            __syncthreads();  // lc reused next chunk
        }

        ts[tid] = t0;
        ts[p1]  = t1;
        __syncthreads();

        // ---- s[j][k] = sum_l w[j][k][l] * t[j][l] ----
#pragma unroll
        for (int q = 0; q < 4; ++q) {
            const int idx = tid + q * 256;   // j*16 + k
            const int j = idx >> 4, k = idx & 15;
            float s = 0.0f;
#pragma unroll
            for (int l = 0; l < L_IN; ++l)
                s = fmaf(ws[(j * K_OUT + k) * L_IN + l], ts[j * L_IN + l], s);
            sb[idx] = s;
        }
        __syncthreads();

        // ---- squash scale per j ----
        if (tid < J_OUT) {
            float sq = 0.0f;
            for (int k = 0; k < K_OUT; ++k) {
                const float s = sb[tid * K_OUT + k];
                sq = fmaf(s, s, sq);
            }
            sc[tid] = (sq / (1.0f + sq)) * rsqrtf(sq + EPS_F);
        }
        __syncthreads();

        // ---- out[j][k] = scale[j] * s[j][k] (in place) ----
#pragma unroll
        for (int q = 0; q < 4; ++q) {
            const int idx = tid + q * 256;
            sb[idx] *= sc[idx >> 4];
        }
        __syncthreads();

        // ---- V[j][l] += sum_k w[j][k][l] * out[j][k] ----
        {
            float a0 = 0.0f, a1 = 0.0f;
#pragma unroll
            for (int k = 0; k < K_OUT; ++k) {
                a0 = fmaf(ws[(j0 * K_OUT + k) * L_IN + l0], sb[j0 * K_OUT + k], a0);
                a1 = fmaf(ws[(j1 * K_OUT + k) * L_IN + l1], sb[j1 * K_OUT + k], a1);
            }
            Vs[tid] += a0;
            Vs[p1]  += a1;
        }
        __syncthreads();
    }

    // ---- write out (N, J, K) ----
    float* o = out + (size_t)n * J_OUT * K_OUT;
#pragma unroll
    for (int q = 0; q < 4; ++q)
        o[tid + q * 256] = sb[tid + q * 256];
}

extern "C" void kernel_launch(void* const* d_in, const int* in_sizes, int n_in,
                              void* d_out, int out_size, void* d_ws, size_t ws_size,
                              hipStream_t stream) {
    const float* x   = (const float*)d_in[0];   // (N, 1152, 8) f32
    const float* w   = (const float*)d_in[1];   // (64, 16, 8)  f32
    float*       out = (float*)d_out;           // (N, 64, 16)  f32

    const int N = in_sizes[0] / (I_IN * L_IN);  // 32
    const size_t smem_floats =
        I_IN * L_IN +            // xs
        J_OUT * K_OUT * L_IN +   // ws
        J_OUT * L_IN +           // Vs
        J_OUT * ITILE +          // lc
        J_OUT * L_IN +           // ts
        J_OUT * K_OUT +          // sb
        J_OUT;                   // sc
    capsule_route_kernel<<<N, 256, smem_floats * sizeof(float), stream>>>(x, w, out);
}